// MoELayer_41721312314327
// MI455X (gfx1250) — compile-verified
//
#include <hip/hip_runtime.h>
#include <hip/hip_bf16.h>
#include <math.h>

#define D_MODEL 1024
#define D_FF    4096
#define N_EXP   8
#define NTOK    4096   // 4 * 1024 tokens

#define KC        32          // K chunk staged per pipeline stage
#define XS_STRIDE 36          // 32 + 4 pad (bank decorrelation)
#define WS_STRIDE 132         // 128 + 4 pad

typedef float v2f __attribute__((ext_vector_type(2)));
typedef float v8f __attribute__((ext_vector_type(8)));
typedef int   v4i __attribute__((ext_vector_type(4)));

typedef __attribute__((address_space(1))) v4i* gptr_v4i;
typedef __attribute__((address_space(3))) v4i* lptr_v4i;

// ---------------- workspace layout ----------------
#define WS_COUNTS 0
#define WS_BUCKET 64
#define WS_GATES  (64 + N_EXP * NTOK * 4)
#define WS_H      147968

// ---------------- async global->LDS staging (CDNA5) ----------------
#if defined(__has_builtin)
#  if __has_builtin(__builtin_amdgcn_global_load_async_to_lds_b128)
#    define USE_ASYNC_LDS 1
#  endif
#endif
#ifndef USE_ASYNC_LDS
#  define USE_ASYNC_LDS 0
#endif

__device__ __forceinline__ void stage16_issue(float* lds, const float* g, float4* reg) {
#if USE_ASYNC_LDS
    __builtin_amdgcn_global_load_async_to_lds_b128(
        (gptr_v4i)g, (lptr_v4i)lds, 0, 0);
    (void)reg;
#else
    *reg = *(const float4*)g;   // global load; s_wait deferred to commit
    (void)lds;
#endif
}
__device__ __forceinline__ void stage16_commit(float* lds, const float4* reg) {
#if !USE_ASYNC_LDS
    *(float4*)lds = *reg;
#else
    (void)lds; (void)reg;
#endif
}
__device__ __forceinline__ void stage_wait() {
#if USE_ASYNC_LDS
    asm volatile("s_wait_asynccnt 0x0" ::: "memory");
#endif
}

// ---------------- kernel 0: zero counters ----------------
__global__ void zero_counts_kernel(int* counts) {
    if (threadIdx.x < N_EXP) counts[threadIdx.x] = 0;
}

// ---------------- kernel 1: router (one wave32 per token) ----------------
__global__ __launch_bounds__(256) void router_kernel(
    const float* __restrict__ x, const float* __restrict__ rW,
    const float* __restrict__ rb, int* __restrict__ counts,
    int* __restrict__ bucket, float* __restrict__ gates)
{
    const int wave  = threadIdx.x >> 5;
    const int lane  = threadIdx.x & 31;
    const int token = blockIdx.x * 8 + wave;
    if (token >= NTOK) return;

    const float* xr = x + (size_t)token * D_MODEL;
    float acc[N_EXP];
#pragma unroll
    for (int e = 0; e < N_EXP; ++e) acc[e] = 0.f;

    for (int d = lane; d < D_MODEL; d += 32) {
        const float xv = xr[d];
        const float* w = rW + (size_t)d * N_EXP;
#pragma unroll
        for (int e = 0; e < N_EXP; ++e) acc[e] += xv * w[e];
    }
#pragma unroll
    for (int e = 0; e < N_EXP; ++e) {
#pragma unroll
        for (int off = 16; off > 0; off >>= 1)
            acc[e] += __shfl_xor(acc[e], off, 32);
    }
    if (lane == 0) {
        float logit[N_EXP];
        float mx = -1e30f;
        int best = 0;
#pragma unroll
        for (int e = 0; e < N_EXP; ++e) {
            logit[e] = acc[e] + rb[e];
            if (logit[e] > mx) { mx = logit[e]; best = e; }
        }
        float sum = 0.f;
#pragma unroll
        for (int e = 0; e < N_EXP; ++e) sum += __expf(logit[e] - mx);
        const float gate = __expf(logit[best] - mx) / sum;  // top-1 prob
        const int pos = atomicAdd(&counts[best], 1);
        bucket[best * NTOK + pos] = token;
        gates[token] = gate;
    }
}

// ---------------- FFN GEMM kernel (templated fc1 / fc2) ---------------
// Workgroup: 32 gathered token rows x 128 output columns, 8 waves.
// Wave w owns cols [cbase+16w, +16) with TWO 16x16 accumulators
// (rows 0-15 and 16-31) sharing each B fragment.
// K pipelined in chunks of 32 through double-buffered LDS.
//
// Fragment layouts (ISA 7.12.2, 32-bit):
//   A: h = lane>>4, m = lane&15 ; a[i] = A[m][2h+i]
//   B: b[i] = B[2h+i][lane&15]
//   C/D: vgpr r -> M = r + 8h, N = lane&15

template <int KTOT, int NTOT, bool IS_FC1>
__global__ __launch_bounds__(256) void ffn_kernel(
    const float* __restrict__ A,     // [NTOK, KTOT] (x or H), gathered by token
    const float* __restrict__ Wm,    // [E, KTOT, NTOT]
    const float* __restrict__ bias,  // [E, NTOT]
    const int* __restrict__ counts,
    const int* __restrict__ bucket,
    const float* __restrict__ gates, // used when !IS_FC1
    float* __restrict__ out)         // [NTOK, NTOT] (H or final out)
{
    const int e    = blockIdx.z;
    const int cnt  = counts[e];
    const int tile = blockIdx.y;
    if (tile * 32 >= cnt) return;
    const int cbase = blockIdx.x * 128;

    const float* W = Wm + (size_t)e * KTOT * NTOT;

    __shared__ float Xs[2][32 * XS_STRIDE];
    __shared__ float Ws[2][KC * WS_STRIDE];
    __shared__ int   toks[32];

    const int tid = threadIdx.x;
    if (tid < 32) {
        const int s = tile * 32 + tid;
        toks[tid] = (s < cnt) ? bucket[e * NTOK + s] : -1;
    }
    __syncthreads();

    // per-thread staging coordinates
    const int xrow = tid >> 3;              // 0..31
    const int xc4  = (tid & 7) * 4;         // 0..28
    const int xtok = toks[xrow];
    const int wr0  = tid >> 5;              // W row base (stride 8 per i)
    const int wc4  = (tid & 31) * 4;        // 0..124

    // zero-fill padded token rows once (both buffers); never rewritten
    if (xtok < 0) {
        const float4 z = make_float4(0.f, 0.f, 0.f, 0.f);
        *(float4*)(&Xs[0][xrow * XS_STRIDE + xc4]) = z;
        *(float4*)(&Xs[1][xrow * XS_STRIDE + xc4]) = z;
    }

    const int wave = tid >> 5;
    const int lane = tid & 31;
    const int h    = lane >> 4;
    const int n    = lane & 15;

    v8f acc0 = {};
    v8f acc1 = {};

    float4 rX;
    float4 rW4[4];

    const int nch = KTOT / KC;

    // ---- prologue: stage chunk 0 into buffer 0 ----
    if (xtok >= 0)
        stage16_issue(&Xs[0][xrow * XS_STRIDE + xc4],
                      A + (size_t)xtok * KTOT + xc4, &rX);
#pragma unroll
    for (int i = 0; i < 4; ++i)
        stage16_issue(&Ws[0][(wr0 + i * 8) * WS_STRIDE + wc4],
                      W + (size_t)(wr0 + i * 8) * NTOT + cbase + wc4, &rW4[i]);
    if (xtok >= 0)
        stage16_commit(&Xs[0][xrow * XS_STRIDE + xc4], &rX);
#pragma unroll
    for (int i = 0; i < 4; ++i)
        stage16_commit(&Ws[0][(wr0 + i * 8) * WS_STRIDE + wc4], &rW4[i]);
    stage_wait();
    __syncthreads();

    // ---- pipelined main loop ----
    for (int c = 0; c < nch; ++c) {
        const int cur = c & 1;
        const int nxt = cur ^ 1;
        const int kc1 = (c + 1) * KC;

        if (c + 1 < nch) {
            if (xtok >= 0)
                stage16_issue(&Xs[nxt][xrow * XS_STRIDE + xc4],
                              A + (size_t)xtok * KTOT + kc1 + xc4, &rX);
#pragma unroll
            for (int i = 0; i < 4; ++i)
                stage16_issue(&Ws[nxt][(wr0 + i * 8) * WS_STRIDE + wc4],
                              W + (size_t)(kc1 + wr0 + i * 8) * NTOT + cbase + wc4,
                              &rW4[i]);
        }

        const float* xs = Xs[cur];
        const float* ws = Ws[cur];
#pragma unroll
        for (int k = 0; k < KC; k += 4) {
            v2f a0, a1, b;
            a0.x = xs[n * XS_STRIDE + k + 2 * h];
            a0.y = xs[n * XS_STRIDE + k + 2 * h + 1];
            a1.x = xs[(n + 16) * XS_STRIDE + k + 2 * h];
            a1.y = xs[(n + 16) * XS_STRIDE + k + 2 * h + 1];
            b.x  = ws[(k + 2 * h)     * WS_STRIDE + wave * 16 + n];
            b.y  = ws[(k + 2 * h + 1) * WS_STRIDE + wave * 16 + n];
            acc0 = __builtin_amdgcn_wmma_f32_16x16x4_f32(
                false, a0, false, b, (short)0, acc0, false, false);
            acc1 = __builtin_amdgcn_wmma_f32_16x16x4_f32(
                false, a1, false, b, (short)0, acc1, false, false);
        }

        if (c + 1 < nch) {
            if (xtok >= 0)
                stage16_commit(&Xs[nxt][xrow * XS_STRIDE + xc4], &rX);
#pragma unroll
            for (int i = 0; i < 4; ++i)
                stage16_commit(&Ws[nxt][(wr0 + i * 8) * WS_STRIDE + wc4], &rW4[i]);
        }
        stage_wait();
        __syncthreads();
    }

    // ---- epilogue ----
    const int col  = cbase + wave * 16 + n;
    const float bs = bias[e * NTOT + col];
#pragma unroll
    for (int r = 0; r < 8; ++r) {
        const int m  = r + 8 * h;
        const int t0 = toks[m];
        const int t1 = toks[m + 16];
        if (IS_FC1) {
            if (t0 >= 0) {
                const float v = acc0[r] + bs;
                out[(size_t)t0 * NTOT + col] = v > 0.f ? v : 0.f;
            }
            if (t1 >= 0) {
                const float v = acc1[r] + bs;
                out[(size_t)t1 * NTOT + col] = v > 0.f ? v : 0.f;
            }
        } else {
            if (t0 >= 0)
                out[(size_t)t0 * NTOT + col] = gates[t0] * (acc0[r] + bs);
            if (t1 >= 0)
                out[(size_t)t1 * NTOT + col] = gates[t1] * (acc1[r] + bs);
        }
    }
}

extern "C" void kernel_launch(void* const* d_in, const int* in_sizes, int n_in,
                              void* d_out, int out_size, void* d_ws, size_t ws_size,
                              hipStream_t stream) {
    const float* x   = (const float*)d_in[0];
    const float* rW  = (const float*)d_in[1];
    const float* rb  = (const float*)d_in[2];
    const float* W1  = (const float*)d_in[3];
    const float* b1  = (const float*)d_in[4];
    const float* W2  = (const float*)d_in[5];
    const float* b2  = (const float*)d_in[6];
    float* out = (float*)d_out;

    char* ws = (char*)d_ws;
    int*   counts = (int*)(ws + WS_COUNTS);
    int*   bucket = (int*)(ws + WS_BUCKET);
    float* gates  = (float*)(ws + WS_GATES);
    float* H      = (float*)(ws + WS_H);

    zero_counts_kernel<<<1, 32, 0, stream>>>(counts);

    router_kernel<<<NTOK / 8, 256, 0, stream>>>(x, rW, rb, counts, bucket, gates);

    dim3 g1(D_FF / 128, NTOK / 32, N_EXP);
    ffn_kernel<D_MODEL, D_FF, true><<<g1, 256, 0, stream>>>(
        x, W1, b1, counts, bucket, nullptr, H);

    dim3 g2(D_MODEL / 128, NTOK / 32, N_EXP);
    ffn_kernel<D_FF, D_MODEL, false><<<g2, 256, 0, stream>>>(
        H, W2, b2, counts, bucket, gates, out);
}